// GCNClassifier_48490180772588
// MI455X (gfx1250) — compile-verified
//
#include <hip/hip_runtime.h>

#define N_NODES 50000
#define N_EDGES 1600000
#define F_INDIM 128
#define HDIM    160
#define G_GR    512
#define C_CLS   2
#define BN_EPS  1e-5f

typedef __attribute__((ext_vector_type(2))) float v2f;
typedef __attribute__((ext_vector_type(8))) float v8f;

// ---------------- degree / normalization ----------------
__global__ __launch_bounds__(256) void k_deg_init(float* deg) {
  int n = blockIdx.x * 256 + threadIdx.x;
  if (n < N_NODES) deg[n] = 1.0f;           // self-loop contributes 1
}

__global__ __launch_bounds__(256) void k_deg_accum(const int* __restrict__ dst, float* deg) {
  int e = blockIdx.x * 256 + threadIdx.x;
  if (e < N_EDGES) atomicAdd(&deg[dst[e]], 1.0f);
}

__global__ __launch_bounds__(256) void k_inv(const float* __restrict__ deg, float* inv) {
  int n = blockIdx.x * 256 + threadIdx.x;
  if (n < N_NODES) {
    float d = deg[n];
    inv[n] = d > 0.0f ? rsqrtf(d) : 0.0f;
  }
}

// ---------------- f32 WMMA GEMM: C[M,Nc] = A[M,K] @ B[K,Nc] (+bias, relu) ----
// One wave per 16x16 output tile; whole B staged in LDS (<=100KB, WGP has 320KB).
// Requires M%16==0, Nc%16==0, K%4==0.
__global__ __launch_bounds__(256) void k_gemm_wmma(
    const float* __restrict__ A, const float* __restrict__ B,
    const float* __restrict__ bias, float* __restrict__ C,
    int M, int K, int Nc, int relu) {
  extern __shared__ float sB[];             // K*Nc floats
  const int tid = (int)threadIdx.x;
  const int total = K * Nc;                 // multiple of 4
  for (int i = tid * 4; i < total; i += 256 * 4)
    *(float4*)(sB + i) = *(const float4*)(B + i);   // coalesced stage
  __syncthreads();

  int wave    = blockIdx.x * 8 + (tid >> 5);
  int tiles_n = Nc >> 4;
  int tiles_m = M >> 4;
  if (wave >= tiles_m * tiles_n) return;    // wave-uniform exit, EXEC all-1 inside
  int tm   = wave / tiles_n;
  int tn   = wave % tiles_n;
  int lane = tid & 31;
  int half = lane >> 4;                     // 0: K pair {0,1}; 1: K pair {2,3}
  int l    = lane & 15;
  int row  = tm * 16 + l;                   // A row handled by this lane
  int col  = tn * 16 + l;                   // B/C column handled by this lane

  v8f acc = {0.f, 0.f, 0.f, 0.f, 0.f, 0.f, 0.f, 0.f};
  const float* ap = A + (long long)row * K + half * 2;
  const float* bp = sB + (half * 2) * Nc + col;     // LDS pointer
  for (int k = 0; k < K; k += 4) {
    v2f a, b;
    a.x = ap[k];                            // A[row][k + 2*half]     (global b64)
    a.y = ap[k + 1];
    b.x = bp[k * Nc];                       // B[k + 2*half][col]     (ds_load)
    b.y = bp[(k + 1) * Nc];
    acc = __builtin_amdgcn_wmma_f32_16x16x4_f32(
        /*neg_a=*/false, a, /*neg_b=*/false, b,
        /*c_mod=*/(short)0, acc, /*reuse_a=*/false, /*reuse_b=*/false);
  }
  float bv = bias ? bias[col] : 0.0f;
  float* cp = C + (long long)(tm * 16 + half * 8) * Nc + col;
#pragma unroll
  for (int i = 0; i < 8; ++i) {             // VGPR i -> row tm*16 + half*8 + i
    float v = acc[i] + bv;
    if (relu) v = fmaxf(v, 0.0f);
    cp[(long long)i * Nc] = v;
  }
}

// ------- out[n][f] = bias[f] + hlin[n][f]*inv[n]^2 ; grid=N, block=HDIM ------
__global__ __launch_bounds__(HDIM) void k_init_self(
    const float* __restrict__ hlin, const float* __restrict__ inv,
    const float* __restrict__ bias, float* __restrict__ out) {
  int n = blockIdx.x;
  int f = threadIdx.x;
  float w = inv[n] * inv[n];                // uniform
  long long i = (long long)n * HDIM + f;
  out[i] = bias[f] + hlin[i] * w;
}

// ------- edge scatter: out[dst] += h[src]*inv[s]*inv[d] ; grid=E, block=HDIM -
__global__ __launch_bounds__(HDIM) void k_aggregate(
    const float* __restrict__ h, const int* __restrict__ src,
    const int* __restrict__ dst, const float* __restrict__ inv,
    float* __restrict__ out) {
  int e = blockIdx.x;
  int f = threadIdx.x;
  int s = src[e], d = dst[e];               // uniform scalar loads
  float w = inv[s] * inv[d];                // uniform
  float v = h[(long long)s * HDIM + f];     // coalesced L2-resident gather
  atomicAdd(out + (long long)d * HDIM + f, v * w);
}

// ------- BN (running stats) + ReLU, in place ; grid=N, block=HDIM ------------
__global__ __launch_bounds__(HDIM) void k_bn_relu(
    float* __restrict__ h, const float* __restrict__ g, const float* __restrict__ be,
    const float* __restrict__ m, const float* __restrict__ v) {
  int f = threadIdx.x;
  long long i = (long long)blockIdx.x * HDIM + f;
  float x = h[i];
  h[i] = fmaxf((x - m[f]) * rsqrtf(v[f] + BN_EPS) * g[f] + be[f], 0.0f);
}

// ---------------- global mean pool -------------------------------------------
__global__ __launch_bounds__(256) void k_pool_init(float* sums, float* cnt) {
  int i = blockIdx.x * 256 + threadIdx.x;
  if (i < G_GR * HDIM) sums[i] = 0.0f;
  if (i < G_GR) cnt[i] = 0.0f;
}

__global__ __launch_bounds__(256) void k_cnt(const int* __restrict__ batch, float* cnt) {
  int n = blockIdx.x * 256 + threadIdx.x;
  if (n < N_NODES) atomicAdd(&cnt[batch[n]], 1.0f);
}

// grid=N, block=HDIM
__global__ __launch_bounds__(HDIM) void k_pool_sum(
    const float* __restrict__ h, const int* __restrict__ batch, float* __restrict__ sums) {
  int n = blockIdx.x;
  int f = threadIdx.x;
  int g = batch[n];                         // uniform
  atomicAdd(sums + (long long)g * HDIM + f, h[(long long)n * HDIM + f]);
}

// grid=G, block=HDIM
__global__ __launch_bounds__(HDIM) void k_pool_div(float* sums, const float* __restrict__ cnt) {
  int g = blockIdx.x;
  int f = threadIdx.x;
  sums[(long long)g * HDIM + f] /= fmaxf(cnt[g], 1.0f);
}

// ---------------- tiny final layer: out = z @ Wc2 + bc2 ----------------------
__global__ __launch_bounds__(256) void k_head2(
    const float* __restrict__ z, const float* __restrict__ W,
    const float* __restrict__ b, float* __restrict__ out) {
  int i = blockIdx.x * 256 + threadIdx.x;
  if (i >= G_GR * C_CLS) return;
  int g = i / C_CLS, c = i % C_CLS;
  float acc = b[c];
  const float* zp = z + g * (HDIM / 2);
#pragma unroll 8
  for (int k = 0; k < HDIM / 2; ++k) acc += zp[k] * W[k * C_CLS + c];
  out[i] = acc;
}

static inline int blks(long long n) { return (int)((n + 255) / 256); }

extern "C" void kernel_launch(void* const* d_in, const int* in_sizes, int n_in,
                              void* d_out, int out_size, void* d_ws, size_t ws_size,
                              hipStream_t stream) {
  const float* x     = (const float*)d_in[0];
  const int*   ei    = (const int*)  d_in[1];
  const int*   batch = (const int*)  d_in[2];
  const float* W1  = (const float*)d_in[3];  const float* b1  = (const float*)d_in[4];
  const float* W2  = (const float*)d_in[5];  const float* b2  = (const float*)d_in[6];
  const float* W3  = (const float*)d_in[7];  const float* b3  = (const float*)d_in[8];
  const float* g1  = (const float*)d_in[9];  const float* be1 = (const float*)d_in[10];
  const float* m1  = (const float*)d_in[11]; const float* v1  = (const float*)d_in[12];
  const float* g2  = (const float*)d_in[13]; const float* be2 = (const float*)d_in[14];
  const float* m2  = (const float*)d_in[15]; const float* v2  = (const float*)d_in[16];
  const float* g3  = (const float*)d_in[17]; const float* be3 = (const float*)d_in[18];
  const float* m3  = (const float*)d_in[19]; const float* v3  = (const float*)d_in[20];
  const float* Wc1 = (const float*)d_in[21]; const float* bc1 = (const float*)d_in[22];
  const float* Wc2 = (const float*)d_in[23]; const float* bc2 = (const float*)d_in[24];
  const int* src = ei;
  const int* dst = ei + N_EDGES;
  float* out = (float*)d_out;

  // workspace carve-up (~65 MB)
  float* ws   = (float*)d_ws;
  float* deg  = ws;
  float* inv  = deg  + N_NODES;
  float* bufA = inv  + N_NODES;                       // GEMM output (h = x@W)
  float* bufB = bufA + (long long)N_NODES * HDIM;     // aggregated features
  float* sums = bufB + (long long)N_NODES * HDIM;     // pooled sums / pooled mean
  float* cnt  = sums + (long long)G_GR * HDIM;
  float* z1   = cnt  + G_GR;                          // head hidden [G,80]

  // degree + inv sqrt (shared by all three layers)
  k_deg_init <<<blks(N_NODES), 256, 0, stream>>>(deg);
  k_deg_accum<<<blks(N_EDGES), 256, 0, stream>>>(dst, deg);
  k_inv      <<<blks(N_NODES), 256, 0, stream>>>(deg, inv);

  // GCN layer: lin (WMMA, B in LDS) -> self-loop+bias init -> scatter -> BN+ReLU
  auto layer = [&](const float* in, int K, const float* W, const float* b,
                   const float* g, const float* be, const float* m, const float* v) {
    int waves  = (N_NODES / 16) * (HDIM / 16);
    int blocks = (waves + 7) / 8;
    size_t lds = (size_t)K * HDIM * sizeof(float);    // <=100KB, fits WGP LDS
    k_gemm_wmma<<<blocks, 256, lds, stream>>>(in, W, nullptr, bufA,
                                              N_NODES, K, HDIM, 0);
    k_init_self<<<N_NODES, HDIM, 0, stream>>>(bufA, inv, b, bufB);
    k_aggregate<<<N_EDGES, HDIM, 0, stream>>>(bufA, src, dst, inv, bufB);
    k_bn_relu  <<<N_NODES, HDIM, 0, stream>>>(bufB, g, be, m, v);
  };

  layer(x,    F_INDIM, W1, b1, g1, be1, m1, v1);
  layer(bufB, HDIM,    W2, b2, g2, be2, m2, v2);
  layer(bufB, HDIM,    W3, b3, g3, be3, m3, v3);

  // global mean pool
  k_pool_init<<<blks(G_GR * HDIM), 256, 0, stream>>>(sums, cnt);
  k_cnt      <<<blks(N_NODES), 256, 0, stream>>>(batch, cnt);
  k_pool_sum <<<N_NODES, HDIM, 0, stream>>>(bufB, batch, sums);
  k_pool_div <<<G_GR, HDIM, 0, stream>>>(sums, cnt);

  // MLP head: [512,160]@[160,80]+bias, ReLU (WMMA), then tiny [512,80]@[80,2]
  {
    int waves  = (G_GR / 16) * ((HDIM / 2) / 16);
    int blocks = (waves + 7) / 8;
    size_t lds = (size_t)HDIM * (HDIM / 2) * sizeof(float);   // 50KB
    k_gemm_wmma<<<blocks, 256, lds, stream>>>(sums, Wc1, bc1, z1,
                                              G_GR, HDIM, HDIM / 2, 1);
  }
  k_head2<<<blks(G_GR * C_CLS), 256, 0, stream>>>(z1, Wc2, bc2, out);
}